// Model_76742475645506
// MI455X (gfx1250) — compile-verified
//
#include <hip/hip_runtime.h>
#include <math.h>

typedef float v2f __attribute__((ext_vector_type(2)));
typedef float v8f __attribute__((ext_vector_type(8)));

#define B_    64
#define L_    1024
#define C_    512
#define P_    720
#define NPAD  768
#define NLEN  512          // IN_LEN
#define KDIM  1024         // 2 * NLEN (seasonal | trend)
#define MDIM  32768        // B_ * C_

// Pair-interleaved layouts (k-pairs adjacent per element -> b64 fragment loads):
//   A2[(k>>1) * 2*MDIM + 2*m + (k&1)]   = A[k][m]
//   W2[(k>>1) * 2*NPAD + 2*p + (k&1)]   = Wt[k][p]

// ---------------------------------------------------------------------------
// Kernel 1: orthonormal DCT-II basis  Mdct[k][n] = s_k * cos(pi*(n+0.5)*k/512)
// phase reduced exactly: (n+0.5)*k is exact in fp32 (< 2^20 half-integers),
// period of cos(pi*x/512) in x is 1024.
// ---------------------------------------------------------------------------
__global__ __launch_bounds__(256) void build_mdct(float* __restrict__ Mdct) {
    int id = blockIdx.x * 256 + threadIdx.x;
    if (id >= NLEN * NLEN) return;
    int k = id >> 9, n = id & 511;
    float scale = (k == 0) ? 0.04419417382415922f /* sqrt(1/512) */
                           : 0.0625f              /* sqrt(2/512) */;
    float ph = fmodf((n + 0.5f) * (float)k, 1024.0f);
    Mdct[id] = scale * cosf(ph * (float)(3.14159265358979323846 / 512.0));
}

// ---------------------------------------------------------------------------
// Kernel 2: fused bias (b_seasonal + b_trend), zero padded to NPAD
// ---------------------------------------------------------------------------
__global__ __launch_bounds__(256) void build_bias(const float* __restrict__ bs,
                                                  const float* __restrict__ bt,
                                                  float* __restrict__ bias) {
    int p = blockIdx.x * 256 + threadIdx.x;
    if (p < NPAD) bias[p] = (p < P_) ? (bs[p] + bt[p]) : 0.0f;
}

// ---------------------------------------------------------------------------
// Kernel 3: fused weights into pair-interleaved B operand:
//   k <  512 : (1/512) * sum_q W_seasonal[p][q] * Mdct[q][k]
//   k >= 512 : (1/512) * sum_q W_trend  [p][q] * Mdct[q][k-512]
// p uniform per block -> scalar loads of the W row; Mdct column reads coalesced.
// grid = (768, 4), 256 threads. p >= 720 columns zero-filled (GEMM padding).
// ---------------------------------------------------------------------------
__global__ __launch_bounds__(256) void build_w(const float* __restrict__ Ws,
                                               const float* __restrict__ Wtr,
                                               const float* __restrict__ Mdct,
                                               float* __restrict__ W2) {
    const int p  = blockIdx.x;                      // 0..767
    const int kk = blockIdx.y * 256 + threadIdx.x;  // 0..1023
    const size_t dst = (size_t)(kk >> 1) * (2 * NPAD) + 2 * p + (kk & 1);
    if (p >= P_) { W2[dst] = 0.0f; return; }
    const int br = kk >> 9;        // 0 = seasonal, 1 = trend
    const int n  = kk & 511;
    const float* __restrict__ wrow = (br == 0 ? Ws : Wtr) + (size_t)p * NLEN;
    const float* __restrict__ mc   = Mdct + n;
    float acc = 0.0f;
    #pragma unroll 4
    for (int q = 0; q < NLEN; ++q)
        acc = fmaf(wrow[q], mc[(size_t)q * NLEN], acc);
    W2[dst] = acc * (1.0f / (float)NLEN);
}

// ---------------------------------------------------------------------------
// Kernel 4: moving-average decomposition (edge-padded window 25) + Haar
// downsample -> pair-interleaved A operand. Each thread owns (b,c), runs a
// sliding-window sum over a 128-long l segment, buffers two consecutive
// downsample outputs and emits coalesced float2 stores.
// grid = (64, 2, 8), 256 threads.
// ---------------------------------------------------------------------------
__global__ __launch_bounds__(256) void decomp(const float* __restrict__ x,
                                              float* __restrict__ A2) {
    const int b  = blockIdx.x;
    const int c  = blockIdx.y * 256 + threadIdx.x;
    const int m  = b * C_ + c;
    const int l0 = blockIdx.z * 128;              // multiple of 4: pairs stay inside
    const float* __restrict__ xb = x + (size_t)b * L_ * C_ + c;

    float S = 0.0f;                               // window sum for center l0
    #pragma unroll
    for (int i = 0; i < 25; ++i) {
        int idx = l0 - 12 + i;
        idx = idx < 0 ? 0 : (idx > L_ - 1 ? L_ - 1 : idx);
        S += xb[(size_t)idx * C_];
    }

    const float inv25 = 1.0f / 25.0f;
    const float is2   = 0.70710678118654752440f;
    float es = 0.0f, et = 0.0f, vs0 = 0.0f, vt0 = 0.0f;
    for (int l = l0; l < l0 + 128; ++l) {
        float tr = S * inv25;
        float xl = xb[(size_t)l * C_];
        float se = xl - tr;
        if ((l & 1) == 0) { es = se; et = tr; }
        else {
            float vs = (es + se) * is2;           // seasonal, index n
            float vt = (et + tr) * is2;           // trend,    index n (+512 in k)
            int n = l >> 1;
            if ((n & 1) == 0) { vs0 = vs; vt0 = vt; }
            else {
                v2f s2 = { vs0, vs };
                v2f t2 = { vt0, vt };
                *(v2f*)(A2 + (size_t)(n >> 1) * (2 * MDIM) + 2 * m)         = s2;
                *(v2f*)(A2 + (size_t)(256 + (n >> 1)) * (2 * MDIM) + 2 * m) = t2;
            }
        }
        int ip = l + 13; if (ip > L_ - 1) ip = L_ - 1;
        int im = l - 12; if (im < 0)      im = 0;
        S += xb[(size_t)ip * C_] - xb[(size_t)im * C_];
    }
}

// ---------------------------------------------------------------------------
// Kernel 5: GEMM  out[m][p] = sum_k A[k][m] * Wt[k][p] + bias[p]
// via v_wmma_f32_16x16x4_f32. Block = 8 waves; wave tile = 32(M) x 64(N):
// 2 A fragments x 4 B fragments = 8 WMMAs per 6 coalesced b64 loads.
// A fragment (16x4): lane r (0-15) rows, K split across lane halves.
// B fragment (4x16): lane r cols,   K split across lane halves.
// Output routed through LDS (pitch 20 -> conflict-free, 16B-aligned), two
// passes (one per 16-row M-subtile); stores are coalesced float4 along C.
// grid = (128, 12), 256 threads.
// ---------------------------------------------------------------------------
__global__ __launch_bounds__(256) void gemm_wmma(const float* __restrict__ A2,
                                                 const float* __restrict__ W2,
                                                 const float* __restrict__ bias,
                                                 float* __restrict__ out) {
    __shared__ float lds[8][64][20];
    const int tid  = threadIdx.x;
    const int w    = tid >> 5;
    const int lane = tid & 31;
    const int hl   = lane >> 4;     // K-half: lanes 0-15 -> K 0,1; 16-31 -> K 2,3
    const int r    = lane & 15;
    const int m0   = blockIdx.x * 256 + w * 32;
    const int p0b  = blockIdx.y * 64;

    v8f accA0 = {}, accA1 = {}, accA2 = {}, accA3 = {};
    v8f accB0 = {}, accB1 = {}, accB2 = {}, accB3 = {};

    const float* __restrict__ Ap = A2 + (size_t)hl * (2 * MDIM) + 2 * (m0 + r);
    const float* __restrict__ Bp = W2 + (size_t)hl * (2 * NPAD) + 2 * (p0b + r);

    #pragma unroll 2
    for (int k0 = 0; k0 < KDIM; k0 += 4) {
        const float* ap = Ap + (size_t)k0 * MDIM;   // (k0>>1) * 2*MDIM
        const float* bp = Bp + (size_t)k0 * NPAD;   // (k0>>1) * 2*NPAD
        v2f a0 = *(const v2f*)(ap);
        v2f a1 = *(const v2f*)(ap + 32);            // second 16-row M-subtile
        v2f b0 = *(const v2f*)(bp);
        v2f b1 = *(const v2f*)(bp + 32);
        v2f b2 = *(const v2f*)(bp + 64);
        v2f b3 = *(const v2f*)(bp + 96);
        accA0 = __builtin_amdgcn_wmma_f32_16x16x4_f32(false, a0, false, b0, (short)0, accA0, false, false);
        accA1 = __builtin_amdgcn_wmma_f32_16x16x4_f32(false, a0, false, b1, (short)0, accA1, false, false);
        accA2 = __builtin_amdgcn_wmma_f32_16x16x4_f32(false, a0, false, b2, (short)0, accA2, false, false);
        accA3 = __builtin_amdgcn_wmma_f32_16x16x4_f32(false, a0, false, b3, (short)0, accA3, false, false);
        accB0 = __builtin_amdgcn_wmma_f32_16x16x4_f32(false, a1, false, b0, (short)0, accB0, false, false);
        accB1 = __builtin_amdgcn_wmma_f32_16x16x4_f32(false, a1, false, b1, (short)0, accB1, false, false);
        accB2 = __builtin_amdgcn_wmma_f32_16x16x4_f32(false, a1, false, b2, (short)0, accB2, false, false);
        accB3 = __builtin_amdgcn_wmma_f32_16x16x4_f32(false, a1, false, b3, (short)0, accB3, false, false);
    }

    const float bp0 = bias[p0b + r];
    const float bp1 = bias[p0b + 16 + r];
    const float bp2 = bias[p0b + 32 + r];
    const float bp3 = bias[p0b + 48 + r];

    const int b  = m0 >> 9;           // m = b*512 + c, 32-row tile stays in one b
    const int c0 = m0 & 511;
    float* __restrict__ ob = out + (size_t)b * (P_ * C_) + c0;

    // ---- pass 0: M-subtile rows m0..m0+15 (accA*) ----
    #pragma unroll
    for (int j = 0; j < 8; ++j) {
        const int ml = j + hl * 8;    // C/D layout: vgpr j -> rows j / j+8
        lds[w][r][ml]      = accA0[j] + bp0;
        lds[w][16 + r][ml] = accA1[j] + bp1;
        lds[w][32 + r][ml] = accA2[j] + bp2;
        lds[w][48 + r][ml] = accA3[j] + bp3;
    }
    __syncthreads();
    #pragma unroll
    for (int it = 0; it < 8; ++it) {
        const int idx = it * 128 + lane * 4;
        const int pl  = idx >> 4;     // 0..63
        const int cl  = idx & 15;     // multiple of 4
        const int p   = p0b + pl;
        if (p < P_) {
            float4 v;
            v.x = lds[w][pl][cl];
            v.y = lds[w][pl][cl + 1];
            v.z = lds[w][pl][cl + 2];
            v.w = lds[w][pl][cl + 3];
            *(float4*)(ob + (size_t)p * C_ + cl) = v;
        }
    }
    __syncthreads();

    // ---- pass 1: M-subtile rows m0+16..m0+31 (accB*) ----
    #pragma unroll
    for (int j = 0; j < 8; ++j) {
        const int ml = j + hl * 8;
        lds[w][r][ml]      = accB0[j] + bp0;
        lds[w][16 + r][ml] = accB1[j] + bp1;
        lds[w][32 + r][ml] = accB2[j] + bp2;
        lds[w][48 + r][ml] = accB3[j] + bp3;
    }
    __syncthreads();
    #pragma unroll
    for (int it = 0; it < 8; ++it) {
        const int idx = it * 128 + lane * 4;
        const int pl  = idx >> 4;
        const int cl  = idx & 15;
        const int p   = p0b + pl;
        if (p < P_) {
            float4 v;
            v.x = lds[w][pl][cl];
            v.y = lds[w][pl][cl + 1];
            v.z = lds[w][pl][cl + 2];
            v.w = lds[w][pl][cl + 3];
            *(float4*)(ob + (size_t)p * C_ + 16 + cl) = v;
        }
    }
}

// ---------------------------------------------------------------------------
extern "C" void kernel_launch(void* const* d_in, const int* in_sizes, int n_in,
                              void* d_out, int out_size, void* d_ws, size_t ws_size,
                              hipStream_t stream) {
    const float* x   = (const float*)d_in[0];
    const float* Ws  = (const float*)d_in[1];
    const float* bs  = (const float*)d_in[2];
    const float* Wtr = (const float*)d_in[3];
    const float* bt  = (const float*)d_in[4];
    float* out = (float*)d_out;

    float* ws_f  = (float*)d_ws;
    float* Mdct  = ws_f;                       // 512*512           = 262144 f
    float* W2    = Mdct + NLEN * NLEN;         // 1024*768          = 786432 f
    float* biasf = W2 + (size_t)KDIM * NPAD;   // 768 f
    float* A2    = biasf + NPAD;               // 1024*32768 f (128 MB)

    build_mdct<<<dim3((NLEN * NLEN + 255) / 256), 256, 0, stream>>>(Mdct);
    build_bias<<<dim3(3), 256, 0, stream>>>(bs, bt, biasf);
    build_w<<<dim3(NPAD, 4), 256, 0, stream>>>(Ws, Wtr, Mdct, W2);
    decomp<<<dim3(B_, C_ / 256, 8), 256, 0, stream>>>(x, A2);
    gemm_wmma<<<dim3(MDIM / 256, NPAD / 64), 256, 0, stream>>>(A2, W2, biasf, out);
}